// CommNet_27127013441953
// MI455X (gfx1250) — compile-verified
//
#include <hip/hip_runtime.h>
#include <hip/hip_bf16.h>

#define HID 256
#define BATCH 512
#define NMODEL 32
#define NAGENT 8
#define NACT 16
#define COMM_W (HID * NAGENT)   // 2048
#define MAX_TILES 64
#define LDS_STRIDE 260

typedef float v2f __attribute__((ext_vector_type(2)));
typedef float v8f __attribute__((ext_vector_type(8)));

// ---------------------------------------------------------------------------
// Kernel A: deterministic grouping of examples by model id.
// paddedPerm[1024]: tile-t slot-r -> batch row (or -1). tileModel[64]: model of
// tile t (or -1 if unused). Stable (rank = #earlier rows with same id): no
// atomics, fully deterministic.
// ---------------------------------------------------------------------------
__global__ __launch_bounds__(BATCH) void build_groups(const int* __restrict__ ids,
                                                      int* __restrict__ paddedPerm,
                                                      int* __restrict__ tileModel) {
    __shared__ int sIds[BATCH];
    __shared__ int sCount[NMODEL];
    __shared__ int sPadOff[NMODEL + 1];
    int t = threadIdx.x;
    sIds[t] = ids[t];
    for (int i = t; i < MAX_TILES * 16; i += BATCH) paddedPerm[i] = -1;
    if (t < MAX_TILES) tileModel[t] = -1;
    __syncthreads();
    int myId = sIds[t];
    int rank = 0;
    for (int j = 0; j < t; ++j) rank += (sIds[j] == myId) ? 1 : 0;
    if (t < NMODEL) {
        int c = 0;
        for (int j = 0; j < BATCH; ++j) c += (sIds[j] == t) ? 1 : 0;
        sCount[t] = c;
    }
    __syncthreads();
    if (t == 0) {
        int off = 0;
        for (int m = 0; m < NMODEL; ++m) {
            sPadOff[m] = off;
            off += ((sCount[m] + 15) >> 4) << 4;
        }
        sPadOff[NMODEL] = off;
    }
    __syncthreads();
    paddedPerm[sPadOff[myId] + rank] = t;
    if (t < NMODEL) {
        for (int tt = sPadOff[t] >> 4; tt < (sPadOff[t + 1] >> 4); ++tt)
            tileModel[tt] = t;
    }
}

// ---------------------------------------------------------------------------
// Kernel B: per 16-row tile of one model: comm-reduce, fused Wc+Wr WMMA GEMM
// chain, tanh -> hidstate; then Wa WMMA + softmax (action) and baseline dots.
// 512 threads = 16 waves; wave w owns output N-tile w (16 cols of 256).
// ---------------------------------------------------------------------------
__global__ __launch_bounds__(BATCH) void hid_kernel(
    const float* __restrict__ comm_in, const int* __restrict__ inp,
    const float* __restrict__ prev_hid,
    const float* __restrict__ Wc, const float* __restrict__ bc,
    const float* __restrict__ Wr, const float* __restrict__ br,
    const float* __restrict__ Wa, const float* __restrict__ ba,
    const float* __restrict__ Wb, const float* __restrict__ bb,
    const float* __restrict__ lut, const float* __restrict__ enc_bias,
    const int* __restrict__ paddedPerm, const int* __restrict__ tileModel,
    float* __restrict__ out_action, float* __restrict__ out_baseline,
    float* __restrict__ out_hid) {
    int tile = blockIdx.x;
    int m = tileModel[tile];
    if (m < 0) return;

    __shared__ float As[16][LDS_STRIDE];
    __shared__ float act[16][NACT + 1];
    __shared__ int rows[16];

    int t = threadIdx.x;
    int lane = t & 31;
    int wave = t >> 5;
    if (t < 16) rows[t] = paddedPerm[tile * 16 + t];
    __syncthreads();

    // Stage A1 = sum over agents of comm_in (zeros for padding rows).
    for (int i = t; i < 16 * HID; i += BATCH) {
        int r = i >> 8, k = i & 255;
        int b = rows[r];
        float s = 0.f;
        if (b >= 0) {
            const float* p = comm_in + ((size_t)b * NAGENT) * HID + k;
#pragma unroll
            for (int a = 0; a < NAGENT; ++a) s += p[(size_t)a * HID];
        }
        As[r][k] = s;
    }
    __syncthreads();

    // WMMA fragment coordinates (16x16x4 f32 layout, ISA 7.12.2).
    int rRow = lane & 15;            // A: M index
    int kOff = (lane >> 4) * 2;      // half-wave K pair select
    int nCol = wave * 16 + rRow;     // B/D: N index for this wave's tile
    int rowHi = (lane >> 4) * 8;     // D: M = v + rowHi

    v8f acc = {0.f, 0.f, 0.f, 0.f, 0.f, 0.f, 0.f, 0.f};
    const float* Wcm = Wc + (size_t)m * HID * HID;
    for (int kk = 0; kk < HID; kk += 4) {
        v2f a, bfrag;
        a.x = As[rRow][kk + kOff];
        a.y = As[rRow][kk + kOff + 1];
        bfrag.x = Wcm[(size_t)(kk + kOff) * HID + nCol];
        bfrag.y = Wcm[(size_t)(kk + kOff + 1) * HID + nCol];
        acc = __builtin_amdgcn_wmma_f32_16x16x4_f32(false, a, false, bfrag,
                                                    (short)0, acc, false, false);
    }
    __syncthreads();

    // Stage A2 = prev_hid; keep accumulating into the same C (fused GEMMs).
    for (int i = t; i < 16 * HID; i += BATCH) {
        int r = i >> 8, k = i & 255;
        int b = rows[r];
        As[r][k] = (b >= 0) ? prev_hid[(size_t)b * HID + k] : 0.f;
    }
    __syncthreads();
    const float* Wrm = Wr + (size_t)m * HID * HID;
    for (int kk = 0; kk < HID; kk += 4) {
        v2f a, bfrag;
        a.x = As[rRow][kk + kOff];
        a.y = As[rRow][kk + kOff + 1];
        bfrag.x = Wrm[(size_t)(kk + kOff) * HID + nCol];
        bfrag.y = Wrm[(size_t)(kk + kOff + 1) * HID + nCol];
        acc = __builtin_amdgcn_wmma_f32_16x16x4_f32(false, a, false, bfrag,
                                                    (short)0, acc, false, false);
    }
    __syncthreads();   // done reading As before overwriting with hid

    // Epilogue: hid = tanh(acc + bc + br + lut[inp[b]] + enc_bias)
    float biasC = bc[(size_t)m * HID + nCol] + br[(size_t)m * HID + nCol] +
                  enc_bias[nCol];
#pragma unroll
    for (int v = 0; v < 8; ++v) {
        int r = v + rowHi;
        int b = rows[r];
        float e = (b >= 0) ? lut[(size_t)inp[b] * HID + nCol] : 0.f;
        float h = tanhf(acc[v] + biasC + e);
        As[r][nCol] = h;
        if (b >= 0) out_hid[(size_t)b * HID + nCol] = h;
    }
    __syncthreads();

    // Action logits: one 16x16 WMMA tile over K=256 (wave 0 only).
    if (wave == 0) {
        v8f aacc = {0.f, 0.f, 0.f, 0.f, 0.f, 0.f, 0.f, 0.f};
        const float* Wam = Wa + (size_t)m * HID * NACT;
        int ncA = lane & 15;
        for (int kk = 0; kk < HID; kk += 4) {
            v2f a, bfrag;
            a.x = As[rRow][kk + kOff];
            a.y = As[rRow][kk + kOff + 1];
            bfrag.x = Wam[(size_t)(kk + kOff) * NACT + ncA];
            bfrag.y = Wam[(size_t)(kk + kOff + 1) * NACT + ncA];
            aacc = __builtin_amdgcn_wmma_f32_16x16x4_f32(false, a, false, bfrag,
                                                         (short)0, aacc, false, false);
        }
        float bav = ba[(size_t)m * NACT + ncA];
#pragma unroll
        for (int v = 0; v < 8; ++v) act[v + rowHi][ncA] = aacc[v] + bav;
    }
    __syncthreads();

    // Softmax per row (16 threads).
    if (t < 16) {
        int b = rows[t];
        if (b >= 0) {
            float mx = act[t][0];
#pragma unroll
            for (int j = 1; j < NACT; ++j) mx = fmaxf(mx, act[t][j]);
            float s = 0.f;
            float e[NACT];
#pragma unroll
            for (int j = 0; j < NACT; ++j) { e[j] = __expf(act[t][j] - mx); s += e[j]; }
            float inv = 1.f / s;
#pragma unroll
            for (int j = 0; j < NACT; ++j)
                out_action[(size_t)b * NACT + j] = e[j] * inv;
        }
    }

    // Baseline: wave w reduces row w (dot(hid, Wb) + bb).
    {
        int r = wave;
        const float* Wbm = Wb + (size_t)m * HID;
        float p = 0.f;
        for (int k = lane; k < HID; k += 32) p += As[r][k] * Wbm[k];
#pragma unroll
        for (int off = 16; off > 0; off >>= 1) p += __shfl_xor(p, off, 32);
        int b = rows[r];
        if (lane == 0 && b >= 0) out_baseline[b] = p + bb[m];
    }
}

// ---------------------------------------------------------------------------
// Kernel C: comm_out = (hid @ Wo[m] + bo[m]) / 7. Grid (tile, colChunk/256);
// 16 waves each own a 16-col WMMA N-tile, K=256.
// ---------------------------------------------------------------------------
__global__ __launch_bounds__(BATCH) void commout_kernel(
    const float* __restrict__ Wo, const float* __restrict__ bo,
    const float* __restrict__ hid,
    const int* __restrict__ paddedPerm, const int* __restrict__ tileModel,
    float* __restrict__ out_comm) {
    int tile = blockIdx.x;
    int m = tileModel[tile];
    if (m < 0) return;
    int cBase = blockIdx.y * 256;

    __shared__ float Hs[16][LDS_STRIDE];
    __shared__ int rows[16];
    int t = threadIdx.x;
    int lane = t & 31;
    int wave = t >> 5;
    if (t < 16) rows[t] = paddedPerm[tile * 16 + t];
    __syncthreads();
    for (int i = t; i < 16 * HID; i += BATCH) {
        int r = i >> 8, k = i & 255;
        int b = rows[r];
        Hs[r][k] = (b >= 0) ? hid[(size_t)b * HID + k] : 0.f;
    }
    __syncthreads();

    int rRow = lane & 15;
    int kOff = (lane >> 4) * 2;
    int nCol = cBase + wave * 16 + rRow;
    int rowHi = (lane >> 4) * 8;

    const float* Wom = Wo + (size_t)m * HID * COMM_W;
    v8f acc = {0.f, 0.f, 0.f, 0.f, 0.f, 0.f, 0.f, 0.f};
    for (int kk = 0; kk < HID; kk += 4) {
        v2f a, bfrag;
        a.x = Hs[rRow][kk + kOff];
        a.y = Hs[rRow][kk + kOff + 1];
        bfrag.x = Wom[(size_t)(kk + kOff) * COMM_W + nCol];
        bfrag.y = Wom[(size_t)(kk + kOff + 1) * COMM_W + nCol];
        acc = __builtin_amdgcn_wmma_f32_16x16x4_f32(false, a, false, bfrag,
                                                    (short)0, acc, false, false);
    }
    float bias = bo[(size_t)m * COMM_W + nCol];
    const float inv7 = 1.0f / 7.0f;
#pragma unroll
    for (int v = 0; v < 8; ++v) {
        int r = v + rowHi;
        int b = rows[r];
        if (b >= 0)
            out_comm[(size_t)b * COMM_W + nCol] = (acc[v] + bias) * inv7;
    }
}

// ---------------------------------------------------------------------------
extern "C" void kernel_launch(void* const* d_in, const int* in_sizes, int n_in,
                              void* d_out, int out_size, void* d_ws, size_t ws_size,
                              hipStream_t stream) {
    const float* comm_in   = (const float*)d_in[0];
    const int*   inp       = (const int*)d_in[1];
    const float* prev_hid  = (const float*)d_in[2];
    // d_in[3] = prev_cell (unused by rnn branch)
    const int*   model_ids = (const int*)d_in[4];
    const float* Wc = (const float*)d_in[5];
    const float* bc = (const float*)d_in[6];
    const float* Wr = (const float*)d_in[7];
    const float* br = (const float*)d_in[8];
    const float* Wa = (const float*)d_in[9];
    const float* ba = (const float*)d_in[10];
    const float* Wb = (const float*)d_in[11];
    const float* bb = (const float*)d_in[12];
    const float* Wo = (const float*)d_in[13];
    const float* bo = (const float*)d_in[14];
    const float* lut      = (const float*)d_in[15];
    const float* enc_bias = (const float*)d_in[16];

    float* out_action   = (float*)d_out;                       // [512,16]
    float* out_baseline = out_action + BATCH * NACT;           // [512,1]
    float* out_hid      = out_baseline + BATCH;                // [512,256]
    float* out_comm     = out_hid + BATCH * HID;               // [512,2048]

    int* paddedPerm = (int*)d_ws;                 // 1024 ints
    int* tileModel  = paddedPerm + MAX_TILES * 16; // 64 ints

    build_groups<<<1, BATCH, 0, stream>>>(model_ids, paddedPerm, tileModel);
    hid_kernel<<<MAX_TILES, BATCH, 0, stream>>>(
        comm_in, inp, prev_hid, Wc, bc, Wr, br, Wa, ba, Wb, bb, lut, enc_bias,
        paddedPerm, tileModel, out_action, out_baseline, out_hid);
    commout_kernel<<<dim3(MAX_TILES, COMM_W / 256), BATCH, 0, stream>>>(
        Wo, bo, out_hid, paddedPerm, tileModel, out_comm);
}